// NanEuclidean_5634997092779
// MI455X (gfx1250) — compile-verified
//
#include <hip/hip_runtime.h>
#include <stdint.h>

// NaN-masked euclidean distances on gfx1250 (MI455X):
//   d[i,j]   = sum_d XX*pY + pX*YY - 2*Xc*Yc
//   cnt[i,j] = sum_d pX*pY
//   out      = sqrt(clip(d,0) / max(1,cnt) * D), NaN where cnt==0
//
// v2: TDM (tensor_load_to_lds) stages X/Y tiles into LDS with hardware row
// padding (stride 34 dwords -> conflict-free, 8B-aligned fragment reads),
// double-buffered and overlapped with compute via TENSORcnt. 8 waves/block:
// block computes 128x64 of output; each wave one 16x64 strip (A reused x4).
// Inner product runs on V_WMMA_F32_16X16X4_F32 (full f32 precision; the
// kernel is bandwidth-bound: ~100MB HBM, ~137 GFLOP, so f32 WMMA is free).

typedef float    v2f __attribute__((ext_vector_type(2)));
typedef float    v8f __attribute__((ext_vector_type(8)));
typedef uint32_t v4u __attribute__((ext_vector_type(4)));
typedef uint32_t v8u __attribute__((ext_vector_type(8)));

#define N_ROWS 4096
#define M_COLS 4096
#define DDIM   1024
#define JT     4          // 16-wide j-tiles per wave (wave strip = 16x64)
#define KC     32         // K columns staged per TDM step (32 dwords = 128B/row)
#define LDSTR  34         // LDS row stride in floats (KC + 2-dword TDM pad)
#define BI     128        // block rows  (8 waves * 16)
#define BJ     64         // block cols  (shared Y tile)

// Issue a TDM 2D tile load: tile_rows x KC floats from a row-major f32 tensor
// (row stride DDIM) into LDS at lds_off, with 2-dword padding after each
// 32-dword row. D# layout per CDNA5 ISA §8 (groups 0/1; 2D -> groups 2/3 NULL).
__device__ __forceinline__ void tdm_load_tile(uint32_t lds_off,
                                              const float* gptr,
                                              uint32_t tile_rows) {
    const uint64_t ga = (uint64_t)(uintptr_t)gptr;
    v4u g0;
    g0[0] = 1u;                                   // count=1 (valid), user mode
    g0[1] = lds_off;                              // lds_addr (bytes)
    g0[2] = (uint32_t)ga;                         // global_addr[31:0]
    g0[3] = ((uint32_t)(ga >> 32) & 0x01FFFFFFu)  // global_addr[56:32]
          | (2u << 30);                           // type=2 ("image")
    v8u g1;
    g1[0] = (2u << 16)                            // data_size = 4 bytes
          | (1u << 20)                            // pad_enable
          | (4u << 22)                            // pad_interval: 32 dwords
          | (1u << 25);                           // pad_amount: 2 dwords
    g1[1] = ((uint32_t)DDIM & 0xFFFFu) << 16;     // tensor_dim0 lo16 (=1024)
    g1[2] = (((uint32_t)DDIM >> 16) & 0xFFFFu)    // tensor_dim0 hi16
          | (((uint32_t)N_ROWS & 0xFFFFu) << 16); // tensor_dim1 lo16 (=4096)
    g1[3] = (((uint32_t)N_ROWS >> 16) & 0xFFFFu)  // tensor_dim1 hi16
          | ((uint32_t)KC << 16);                 // tile_dim0 = KC elements
    g1[4] = tile_rows;                            // tile_dim1 (tile_dim2 = 0)
    g1[5] = (uint32_t)DDIM;                       // tensor_dim0_stride lo32
    g1[6] = 0u;                                   // stride hi | dim1_stride lo
    g1[7] = 0u;
    asm volatile("tensor_load_to_lds %0, %1" :: "s"(g0), "s"(g1) : "memory");
}

__global__ __launch_bounds__(256) void nan_euclid_wmma_tdm(
    const float* __restrict__ X,
    const float* __restrict__ Y,
    float* __restrict__ out)
{
    __shared__ float sX[2][BI * LDSTR];
    __shared__ float sY[2][BJ * LDSTR];

    const int lane = threadIdx.x & 31;
    const int wv   = threadIdx.x >> 5;   // 0..7
    const int ln   = lane & 15;          // fragment row (A:M, B:N)
    const int hi   = lane >> 4;          // K-pair selector / M+8 in C

    const int i_base = blockIdx.y * BI;
    const int j_base = blockIdx.x * BJ;

    v8f acc_d[JT] = {};
    v8f acc_c[JT] = {};

    const bool issuer = (threadIdx.x < 32);   // wave 0 drives the TDM

    if (issuer) {
        tdm_load_tile((uint32_t)(uintptr_t)&sX[0][0],
                      X + (size_t)i_base * DDIM, BI);
        tdm_load_tile((uint32_t)(uintptr_t)&sY[0][0],
                      Y + (size_t)j_base * DDIM, BJ);
    }

    int cur = 0;
    for (int k = 0; k < DDIM; k += KC) {
        if (issuer) {
            if (k + KC < DDIM) {
                // Prefetch next step into the other buffer, then retire the
                // current buffer's two loads (leave the 2 new ones in flight).
                tdm_load_tile((uint32_t)(uintptr_t)&sX[cur ^ 1][0],
                              X + (size_t)i_base * DDIM + (k + KC), BI);
                tdm_load_tile((uint32_t)(uintptr_t)&sY[cur ^ 1][0],
                              Y + (size_t)j_base * DDIM + (k + KC), BJ);
                __builtin_amdgcn_s_wait_tensorcnt(2);
            } else {
                __builtin_amdgcn_s_wait_tensorcnt(0);
            }
        }
        __syncthreads();

        const float* xb = &sX[cur][(wv * 16 + ln) * LDSTR + 2 * hi];
        const float* yb = &sY[cur][ln * LDSTR + 2 * hi];

#pragma unroll
        for (int kk = 0; kk < KC; kk += 4) {
            // ---- A side (shared across the wave's 4 j-tiles) ----
            v2f xv = *(const v2f*)(xb + kk);
            const float x0 = xv.x, x1 = xv.y;
            const bool mx0 = __builtin_isnan(x0);
            const bool mx1 = __builtin_isnan(x1);
            const float xc0 = mx0 ? 0.0f : x0;
            const float xc1 = mx1 ? 0.0f : x1;
            v2f a_xx  = { xc0 * xc0, xc1 * xc1 };
            v2f a_px  = { mx0 ? 0.0f : 1.0f, mx1 ? 0.0f : 1.0f };
            v2f a_m2x = { -2.0f * xc0, -2.0f * xc1 };

#pragma unroll
            for (int t = 0; t < JT; ++t) {
                v2f yv = *(const v2f*)(yb + t * 16 * LDSTR + kk);
                const float y0 = yv.x, y1 = yv.y;
                const bool my0 = __builtin_isnan(y0);
                const bool my1 = __builtin_isnan(y1);
                const float yc0 = my0 ? 0.0f : y0;
                const float yc1 = my1 ? 0.0f : y1;
                v2f b_py = { my0 ? 0.0f : 1.0f, my1 ? 0.0f : 1.0f };
                v2f b_yy = { yc0 * yc0, yc1 * yc1 };
                v2f b_yc = { yc0, yc1 };

                acc_d[t] = __builtin_amdgcn_wmma_f32_16x16x4_f32(
                    false, a_xx,  false, b_py, (short)0, acc_d[t], false, false);
                acc_d[t] = __builtin_amdgcn_wmma_f32_16x16x4_f32(
                    false, a_px,  false, b_yy, (short)0, acc_d[t], false, false);
                acc_d[t] = __builtin_amdgcn_wmma_f32_16x16x4_f32(
                    false, a_m2x, false, b_yc, (short)0, acc_d[t], false, false);
                acc_c[t] = __builtin_amdgcn_wmma_f32_16x16x4_f32(
                    false, a_px,  false, b_py, (short)0, acc_c[t], false, false);
            }
        }
        __syncthreads();
        cur ^= 1;
    }

    // ---- Epilogue: C/D layout -> VGPR r holds row M = r + 8*hi, col N = ln
#pragma unroll
    for (int t = 0; t < JT; ++t) {
        const int col = j_base + t * 16 + ln;
#pragma unroll
        for (int r = 0; r < 8; ++r) {
            float d = fmaxf(acc_d[t][r], 0.0f);
            float c = acc_c[t][r];
            float res;
            if (c == 0.0f) {
                res = __builtin_nanf("");
            } else {
                res = __builtin_sqrtf(d / fmaxf(c, 1.0f) * (float)DDIM);
            }
            const int row = i_base + wv * 16 + r + 8 * hi;
            out[(size_t)row * M_COLS + col] = res;
        }
    }
}

extern "C" void kernel_launch(void* const* d_in, const int* in_sizes, int n_in,
                              void* d_out, int out_size, void* d_ws, size_t ws_size,
                              hipStream_t stream) {
    (void)in_sizes; (void)n_in; (void)out_size; (void)d_ws; (void)ws_size;
    const float* X = (const float*)d_in[0];   // [4096, 1024] f32 (with NaNs)
    const float* Y = (const float*)d_in[1];   // [4096, 1024] f32 (with NaNs)
    float* out = (float*)d_out;               // [4096, 4096] f32

    dim3 grid(M_COLS / BJ, N_ROWS / BI);      // (64, 32)
    dim3 block(256);                          // 8 waves: 128x64 output block
    nan_euclid_wmma_tdm<<<grid, block, 0, stream>>>(X, Y, out);
}